// MultiheadChannelAttention_32959579029561
// MI455X (gfx1250) — compile-verified
//
#include <hip/hip_runtime.h>
#include <hip/hip_bf16.h>
#include <math.h>

typedef __attribute__((ext_vector_type(2))) float v2f;
typedef __attribute__((ext_vector_type(8))) float v8f;

#define BATCH   4
#define C_DIM   256
#define HEADS   8
#define DHEAD   2048
#define HH      128
#define WW      128
#define HW      16384           // HH*WW = HEADS*DHEAD

#define CB      32              // S row-block per workgroup
#define KT1     16              // phase-1 K staging chunk (k columns)
#define KSTRIDE 20              // padded LDS row stride (floats), keeps 16B align
#define SSTRIDE 260             // padded S stride (256 would be 16-way bank conflict)
#define KT3     64              // phase-3 V staging chunk (rows)
#define VSTRIDE 68              // padded V stage stride
#define STAGEF  (C_DIM * KSTRIDE + CB * KSTRIDE)   // 5760 floats per stage buffer

// ---------------------------------------------------------------------------
// CDNA5 async global->LDS staging (ASYNCcnt-tracked, no VGPR round trip).
// Builtin signature (from clang diagnostic): arg0 = AS1 pointer to
// vector_size(16)/vector_size(8) int, arg1 = LDS pointer, then imm offset,
// imm cpol. Falls back to synchronous copies if unavailable.
// ---------------------------------------------------------------------------
#if __has_builtin(__builtin_amdgcn_global_load_async_to_lds_b128) && \
    __has_builtin(__builtin_amdgcn_global_load_async_to_lds_b64)
#define ASYNC_STAGING 1
typedef int v4i __attribute__((vector_size(4 * sizeof(int))));
typedef int v2i __attribute__((vector_size(2 * sizeof(int))));
typedef __attribute__((address_space(1))) v4i* g4ptr;
typedef __attribute__((address_space(3))) v4i* l4ptr;
typedef __attribute__((address_space(1))) v2i* g2ptr;
typedef __attribute__((address_space(3))) v2i* l2ptr;
__device__ __forceinline__ void cp_b128(float* lds, const float* g) {
    __builtin_amdgcn_global_load_async_to_lds_b128((g4ptr)g, (l4ptr)lds, 0, 0);
}
__device__ __forceinline__ void cp_b64(float* lds, const float* g) {
    __builtin_amdgcn_global_load_async_to_lds_b64((g2ptr)g, (l2ptr)lds, 0, 0);
}
#if __has_builtin(__builtin_amdgcn_s_wait_asynccnt)
#define WAIT_ASYNC(n) __builtin_amdgcn_s_wait_asynccnt(n)
#else
#define WAIT_ASYNC(n) asm volatile("s_wait_asynccnt %0" :: "n"(n) : "memory")
#endif
#else
#define ASYNC_STAGING 0
__device__ __forceinline__ void cp_b128(float* lds, const float* g) {
    *(float4*)lds = *(const float4*)g;
}
__device__ __forceinline__ void cp_b64(float* lds, const float* g) {
    *(float2*)lds = *(const float2*)g;
}
#define WAIT_ASYNC(n) ((void)0)
#endif

// ---------------------------------------------------------------------------
// xp = x + channel positional encoding; pe is batch-invariant so compute once
// per (c,pos) and broadcast over the 4 batches.
// ---------------------------------------------------------------------------
__global__ void pe_add_kernel(const float* __restrict__ x, float* __restrict__ xp) {
    int idx = blockIdx.x * blockDim.x + threadIdx.x;    // over C*HW
    int c   = idx >> 14;                                 // / 16384
    int pos = idx & (HW - 1);
    int p   = pos >> 1;
    float ang = (float)c * __expf((float)(2 * p) * (-9.210340371976184f / (float)HW));
    float pe  = (pos & 1) ? cosf(ang) : sinf(ang);
    const size_t stride = (size_t)C_DIM * HW;
    size_t o = (size_t)idx;
#pragma unroll
    for (int b = 0; b < BATCH; ++b)
        xp[(size_t)b * stride + o] = x[(size_t)b * stride + o] + pe;
}

// ---------------------------------------------------------------------------
// Depthwise 9x9 conv, same padding. 16x16 output tile per 256-thread block,
// 24x24 halo staged in LDS (stride 25), weights (81) staged in LDS.
// ---------------------------------------------------------------------------
__global__ void dwconv9_kernel(const float* __restrict__ in, const float* __restrict__ wgt,
                               const float* __restrict__ bias, float* __restrict__ out) {
    __shared__ float sT[24 * 25];
    __shared__ float sW[81];
    const int c = blockIdx.y;
    const int b = blockIdx.z;
    const int tileIdx = blockIdx.x;            // 0..63 -> 8x8 tiles
    const int tx = (tileIdx & 7) * 16;
    const int ty = (tileIdx >> 3) * 16;
    const int t  = threadIdx.x;                // 0..255
    const float* plane = in + ((size_t)(b * C_DIM + c)) * HW;

    for (int i = t; i < 576; i += 256) {
        int r  = i / 24, cc = i - r * 24;
        int gy = ty - 4 + r, gx = tx - 4 + cc;
        float v = 0.0f;
        if (gy >= 0 && gy < HH && gx >= 0 && gx < WW) v = plane[gy * WW + gx];
        sT[r * 25 + cc] = v;
    }
    if (t < 81) sW[t] = wgt[c * 81 + t];
    __syncthreads();

    const int lx = t & 15, ly = t >> 4;
    float acc = bias[c];
#pragma unroll
    for (int ky = 0; ky < 9; ++ky)
#pragma unroll
        for (int kx = 0; kx < 9; ++kx)
            acc = fmaf(sW[ky * 9 + kx], sT[(ly + ky) * 25 + (lx + kx)], acc);
    out[((size_t)(b * C_DIM + c)) * HW + (ty + ly) * WW + (tx + lx)] = acc;
}

// ---------------------------------------------------------------------------
// Channel attention for one (b, h, 32-row block of C):
//   phase 1: S = Q K^T / 128   (fp32 WMMA 16x16x4, K=2048, double-buffered
//            async K/Q staging: 5 uniform async ops per thread per chunk)
//   phase 2: row softmax on S (32x256) in LDS
//   phase 3: A = P V           (fp32 WMMA, K=256, N=2048, double-buffered
//            async V staging: 4 uniform async ops per thread per chunk)
// Fragment layouts per CDNA5 ISA 7.12.2:
//   A (16x4 f32): lane l holds M=l&15, K=(l>>4)*2 .. +1 in 2 VGPRs
//   B (4x16 f32): lane l holds N=l&15, K=(l>>4)*2 .. +1 in 2 VGPRs
//   C/D (16x16 f32): vgpr r, lane l -> M = r + (l>>4)*8, N = l&15
// ---------------------------------------------------------------------------
__global__ void __launch_bounds__(256) chan_attn_kernel(
        const float* __restrict__ q, const float* __restrict__ k,
        const float* __restrict__ v, float* __restrict__ aout) {
    // 32*260 (S) + 2 * 5760 (double-buffered stages) = 19840 floats = 79,360 B
    __shared__ __align__(16) float smem[CB * SSTRIDE + 2 * STAGEF];
    float* sS     = smem;
    float* stage0 = smem + CB * SSTRIDE;
    float* stage1 = stage0 + STAGEF;

    const int t  = threadIdx.x;
    const int w  = t >> 5;                       // wave id, 0..7 (wave32)
    const int l  = t & 31;                       // lane
    const int lm = l & 15;
    const int kb = (l >> 4) << 1;                // 0 or 2: lane's K sub-offset
    const int c0 = blockIdx.x * CB;
    const int h  = blockIdx.y;
    const int b  = blockIdx.z;

    const size_t rowStride = (size_t)HEADS * DHEAD;   // 16384 floats between channels
    const float* qbase = q + (size_t)b * C_DIM * rowStride + (size_t)h * DHEAD;
    const float* kbase = k + (size_t)b * C_DIM * rowStride + (size_t)h * DHEAD;
    const float* vbase = v + (size_t)b * C_DIM * rowStride + (size_t)h * DHEAD;
    float*       abase = aout + (size_t)b * C_DIM * rowStride + (size_t)h * DHEAD;

    const int rt = w >> 2;                       // row tile 0..1 (16 rows each)
    const int cq = w & 3;                        // phase1: col quarter; phase3: col tile

    // ---- stage issue helpers (uniform op count across all 256 threads) ----
    auto issue_qk = [&](int chunk, float* buf) {
        float* bK = buf;
        float* bQ = buf + C_DIM * KSTRIDE;
        const int k0 = chunk * KT1;
#pragma unroll
        for (int i = 0; i < 4; ++i) {            // K chunk: 256 rows x 16 cols
            int ii  = i * 256 + t;
            int row = ii >> 2;
            int c4  = (ii & 3) << 2;
            cp_b128(&bK[row * KSTRIDE + c4], kbase + (size_t)row * rowStride + k0 + c4);
        }
        int qr = t >> 3, qc = (t & 7) << 1;      // Q chunk: 32 rows x 16 cols (b64 each)
        cp_b64(&bQ[qr * KSTRIDE + qc], qbase + (size_t)(c0 + qr) * rowStride + k0 + qc);
    };
    auto issue_v = [&](int step, float* buf) {
        const int cb2 = step >> 2;               // 64-col block of DHEAD
        const int kc  = step & 3;                // 64-row chunk of C
#pragma unroll
        for (int i = 0; i < 4; ++i) {            // V chunk: 64 rows x 64 cols
            int ii  = i * 256 + t;
            int row = ii >> 4;
            int c4  = (ii & 15) << 2;
            cp_b128(&buf[row * VSTRIDE + c4],
                    vbase + (size_t)(kc * KT3 + row) * rowStride + cb2 * 64 + c4);
        }
    };

    v8f acc0 = {}, acc1 = {}, acc2 = {}, acc3 = {};
    const v8f vzero = {};

    // ---------------- phase 1: S = Q K^T ----------------
    constexpr int NCH1 = DHEAD / KT1;            // 128 chunks
    issue_qk(0, stage0);
    for (int ci = 0; ci < NCH1; ++ci) {
        float* cur = (ci & 1) ? stage1 : stage0;
        float* nxt = (ci & 1) ? stage0 : stage1;
        if (ci + 1 < NCH1) { issue_qk(ci + 1, nxt); WAIT_ASYNC(5); }
        else               { WAIT_ASYNC(0); }
        __syncthreads();                         // chunk ci resident for all waves
        const float* bK = cur;
        const float* bQ = cur + C_DIM * KSTRIDE;
#pragma unroll
        for (int kk = 0; kk < KT1; kk += 4) {
            v2f a  = *(const v2f*)&bQ[(rt * 16 + lm) * KSTRIDE + kk + kb];
            v2f b0 = *(const v2f*)&bK[(cq * 64 +  0 + lm) * KSTRIDE + kk + kb];
            v2f b1 = *(const v2f*)&bK[(cq * 64 + 16 + lm) * KSTRIDE + kk + kb];
            v2f b2 = *(const v2f*)&bK[(cq * 64 + 32 + lm) * KSTRIDE + kk + kb];
            v2f b3 = *(const v2f*)&bK[(cq * 64 + 48 + lm) * KSTRIDE + kk + kb];
            acc0 = __builtin_amdgcn_wmma_f32_16x16x4_f32(false, a, false, b0, (short)0, acc0, false, false);
            acc1 = __builtin_amdgcn_wmma_f32_16x16x4_f32(false, a, false, b1, (short)0, acc1, false, false);
            acc2 = __builtin_amdgcn_wmma_f32_16x16x4_f32(false, a, false, b2, (short)0, acc2, false, false);
            acc3 = __builtin_amdgcn_wmma_f32_16x16x4_f32(false, a, false, b3, (short)0, acc3, false, false);
        }
        __syncthreads();                         // protect buffer about to be re-filled
    }

    // scale by 1/sqrt(hw)=1/128 and scatter into sS
    const float scale = 1.0f / 128.0f;
#pragma unroll
    for (int r = 0; r < 8; ++r) {
        int m    = r + (kb << 2);                // r + (l>>4)*8
        int srow = (rt * 16 + m) * SSTRIDE;
        sS[srow + cq * 64 +  0 + lm] = acc0[r] * scale;
        sS[srow + cq * 64 + 16 + lm] = acc1[r] * scale;
        sS[srow + cq * 64 + 32 + lm] = acc2[r] * scale;
        sS[srow + cq * 64 + 48 + lm] = acc3[r] * scale;
    }
    __syncthreads();

    // ---------------- phase 2: row softmax ----------------
    {
        int row = t >> 3;                        // 8 lanes per row
        float* rp = &sS[row * SSTRIDE + (t & 7) * 32];
        float mx = -3.4e38f;
#pragma unroll
        for (int j = 0; j < 32; ++j) mx = fmaxf(mx, rp[j]);
        mx = fmaxf(mx, __shfl_xor(mx, 1, 32));
        mx = fmaxf(mx, __shfl_xor(mx, 2, 32));
        mx = fmaxf(mx, __shfl_xor(mx, 4, 32));
        float sum = 0.0f;
#pragma unroll
        for (int j = 0; j < 32; ++j) { float e = __expf(rp[j] - mx); rp[j] = e; sum += e; }
        sum += __shfl_xor(sum, 1, 32);
        sum += __shfl_xor(sum, 2, 32);
        sum += __shfl_xor(sum, 4, 32);
        float inv = 1.0f / sum;
#pragma unroll
        for (int j = 0; j < 32; ++j) rp[j] *= inv;
    }
    __syncthreads();

    // ---------------- phase 3: A = P V ----------------
    constexpr int NST = (DHEAD / 64) * (C_DIM / KT3);   // 32 col-blocks x 4 k-chunks
    v8f o = vzero;
    issue_v(0, stage0);
    for (int st = 0; st < NST; ++st) {
        float* cur = (st & 1) ? stage1 : stage0;
        float* nxt = (st & 1) ? stage0 : stage1;
        if (st + 1 < NST) { issue_v(st + 1, nxt); WAIT_ASYNC(4); }
        else              { WAIT_ASYNC(0); }
        __syncthreads();
        const int cb2 = st >> 2;
        const int kc  = st & 3;
        const int k0  = kc * KT3;
#pragma unroll
        for (int kk = 0; kk < KT3; kk += 4) {
            v2f a = *(const v2f*)&sS[(rt * 16 + lm) * SSTRIDE + k0 + kk + kb];
            v2f bf;
            bf.x = cur[(kk + kb)     * VSTRIDE + cq * 16 + lm];
            bf.y = cur[(kk + kb + 1) * VSTRIDE + cq * 16 + lm];
            o = __builtin_amdgcn_wmma_f32_16x16x4_f32(false, a, false, bf, (short)0, o, false, false);
        }
        if (kc == 3) {                           // finished K sweep for this col block
#pragma unroll
            for (int r = 0; r < 8; ++r) {
                int m = r + (kb << 2);
                abase[(size_t)(c0 + rt * 16 + m) * rowStride + cb2 * 64 + cq * 16 + lm] = o[r];
            }
            o = vzero;
        }
        __syncthreads();                         // protect buffer about to be re-filled
    }
}

// ---------------------------------------------------------------------------
extern "C" void kernel_launch(void* const* d_in, const int* in_sizes, int n_in,
                              void* d_out, int out_size, void* d_ws, size_t ws_size,
                              hipStream_t stream) {
    (void)in_sizes; (void)n_in; (void)out_size; (void)ws_size;
    const float* x   = (const float*)d_in[0];
    const float* q_w = (const float*)d_in[1];
    const float* q_b = (const float*)d_in[2];
    const float* k_w = (const float*)d_in[3];
    const float* k_b = (const float*)d_in[4];
    const float* v_w = (const float*)d_in[5];
    const float* v_b = (const float*)d_in[6];
    const float* o_w = (const float*)d_in[7];
    const float* o_b = (const float*)d_in[8];
    float* out = (float*)d_out;
    float* ws  = (float*)d_ws;

    const size_t N = (size_t)BATCH * C_DIM * HW;   // 16,777,216 floats per tensor
    float* xp = ws;             // x + pe
    float* qq = ws + N;
    float* kk = ws + 2 * N;
    float* vv = ws + 3 * N;
    float* aa = ws;             // attention output reuses xp slot

    pe_add_kernel<<<(C_DIM * HW) / 256, 256, 0, stream>>>(x, xp);

    dim3 cgrid(64, C_DIM, BATCH);                  // 8x8 tiles per plane
    dwconv9_kernel<<<cgrid, 256, 0, stream>>>(xp, q_w, q_b, qq);
    dwconv9_kernel<<<cgrid, 256, 0, stream>>>(xp, k_w, k_b, kk);
    dwconv9_kernel<<<cgrid, 256, 0, stream>>>(xp, v_w, v_b, vv);

    dim3 agrid(C_DIM / CB, HEADS, BATCH);          // 8 x 8 x 4 workgroups
    chan_attn_kernel<<<agrid, 256, 0, stream>>>(qq, kk, vv, aa);

    dwconv9_kernel<<<cgrid, 256, 0, stream>>>(aa, o_w, o_b, out);
}